// HRNN_encoder_5325759447102
// MI455X (gfx1250) — compile-verified
//
#include <hip/hip_runtime.h>
#include <hip/hip_bf16.h>

typedef __attribute__((ext_vector_type(2))) float v2f;
typedef __attribute__((ext_vector_type(8))) float v8f;

#define SEQ_    96
#define B_      64
#define D_      512
#define HID_    256
#define FKD_    128
#define NSUM_   384     // HID_ + FKD_
#define DEPTH_  6
#define HSTR_   516     // LDS row stride for H (pad: 516 % 64 == 4 -> conflict-free frag loads)
#define SSTR_   388     // LDS row stride for S

// output regions (flat float offsets in d_out)
#define O_X     0
#define O_OUT   1572864
#define O_ALLH  1589248
#define O_FK    20463616
#define O_FKC   20500480

__device__ __forceinline__ float hsig(float z) {
    return fminf(fmaxf(0.2f * z + 0.5f, 0.f), 1.f);
}

// ---------------------------------------------------------------------------
// Pack [U | U_FK_1] and [W | W_FK_1] (512 x 384 each) into WMMA-B fragment
// order: packed[((nt*128 + kq)*32 + lane)*2 + j] = M[kq*4 + (lane>>4)*2 + j][nt*16 + (lane&15)]
// so the hot GEMM loops do one coalesced v2f load per lane per K-step.
// ---------------------------------------------------------------------------
__global__ void pack_weights(const float* __restrict__ U, const float* __restrict__ UFK,
                             const float* __restrict__ W, const float* __restrict__ WFK,
                             float* __restrict__ Pu, float* __restrict__ Pw) {
    int p = blockIdx.x * blockDim.x + threadIdx.x;
    const int N = NSUM_ * D_;   // 196608 per packed matrix
    if (p >= 2 * N) return;
    const float* A0;
    const float* A1;
    float* dst;
    int q = p;
    if (p < N) { A0 = U; A1 = UFK; dst = Pu; }
    else       { A0 = W; A1 = WFK; dst = Pw; q = p - N; }
    int j    = q & 1;
    int lane = (q >> 1) & 31;
    int kq   = (q >> 6) & 127;
    int nt   = q >> 13;                       // 0..23
    int n    = nt * 16 + (lane & 15);
    int k    = kq * 4 + ((lane >> 4) << 1) + j;
    dst[q] = (n < HID_) ? A0[k * HID_ + n] : A1[k * FKD_ + (n - HID_)];
}

// ---------------------------------------------------------------------------
// Build X0 = concat([x, zeros]) transposed to (SEQ, B, 512), and copy x -> out
// ---------------------------------------------------------------------------
__global__ void build_x0(const float* __restrict__ x, float* __restrict__ X0,
                         float* __restrict__ outx) {
    int i = blockIdx.x * blockDim.x + threadIdx.x;
    const int total = SEQ_ * B_ * D_;         // 3145728
    if (i < total) {
        int c  = i & (D_ - 1);
        int tb = i >> 9;
        int b  = tb & 63;
        int t  = tb >> 6;
        X0[i] = (c < HID_) ? x[(b * SEQ_ + t) * 256 + c] : 0.f;
    }
    if (i < B_ * SEQ_ * 256) outx[i] = x[i];  // passthrough output #0
}

// ---------------------------------------------------------------------------
// Batched x-GEMM for one layer: XS(6144 x 384) = Xin(6144 x 512) @ [W|W_FK_1]
// One 16x16 tile per wave; 384 M-tiles x 24 N-tiles = 9216 waves.
// ---------------------------------------------------------------------------
__global__ __launch_bounds__(256)
void gemm_x(const float* __restrict__ Xin, const float* __restrict__ Pw,
            float* __restrict__ XS) {
    int l    = threadIdx.x & 31;
    int wave = (blockIdx.x * blockDim.x + threadIdx.x) >> 5;
    int mt   = wave / 24;
    int nt   = wave - mt * 24;
    if (mt >= 384) return;
    int m0 = mt * 16;
    const float* arow = Xin + (m0 + (l & 15)) * D_ + ((l >> 4) << 1);
    const v2f*  bp    = (const v2f*)(Pw + nt * 8192) + l;
    v8f c = {};
#pragma unroll 4
    for (int kq = 0; kq < 128; ++kq) {
        v2f a = *(const v2f*)(arow + kq * 4);
        v2f b = bp[kq * 32];
        c = __builtin_amdgcn_wmma_f32_16x16x4_f32(false, a, false, b, (short)0, c, false, false);
    }
    int n     = nt * 16 + (l & 15);
    int mbase = m0 + ((l >> 4) << 3);
#pragma unroll
    for (int v = 0; v < 8; ++v)
        XS[(mbase + v) * NSUM_ + n] = c[v];
}

// ---------------------------------------------------------------------------
// Sequential scan for one layer. Single 512-thread workgroup (16 wave32).
// LDS: H (h_tm1, 64x516), S (h@[U|U_FK_1], 64x388), hv_tm1 (64).
// Per t: 96 WMMA tiles (6 per wave), then row-phase (8 lanes per row).
// X is updated in place (each element read-then-written by its owner thread).
// ---------------------------------------------------------------------------
__global__ __launch_bounds__(512)
void scan_layer(float* __restrict__ X, const float* __restrict__ XS,
                const float* __restrict__ Pu, const float* __restrict__ mask,
                const float* __restrict__ bvec, const float* __restrict__ bfk1,
                const float* __restrict__ aR, const float* __restrict__ aT,
                const float* __restrict__ wfk2, const float* __restrict__ bfk2,
                const float* __restrict__ gammas, const float* __restrict__ betas,
                const float* __restrict__ FKprev, const float* __restrict__ HVprev,
                float* __restrict__ FKcur, float* __restrict__ HVcur,
                float* __restrict__ out, int layer) {
    extern __shared__ float smem[];
    float* Hs  = smem;                    // 64 x HSTR_
    float* Ss  = smem + B_ * HSTR_;       // 64 x SSTR_
    float* hvs = Ss + B_ * SSTR_;         // 64

    const int tid = threadIdx.x;
    for (int i = tid; i < B_ * HSTR_; i += 512) Hs[i] = 0.f;
    if (tid < 64) hvs[tid] = 0.f;
    __syncthreads();

    const int l   = tid & 31;
    const int w   = tid >> 5;
    const int row = tid >> 3;
    const int j   = tid & 7;
    const float mask3_t = (layer == DEPTH_ - 1) ? 1.f : 0.f;

    float* outALL = out + O_ALLH;
    float* outFK  = out + O_FK;
    float* outFKC = out + O_FKC;
    float* outO   = out + O_OUT;

    for (int t = 0; t < SEQ_; ++t) {
        // ---- GEMM phase: S = H @ [U | U_FK_1] ---------------------------
#pragma unroll
        for (int rep = 0; rep < 6; ++rep) {
            int tile = w + rep * 16;          // 0..95
            int mt   = tile & 3;
            int nt   = tile >> 2;             // 0..23
            int m0   = mt * 16;
            const float* arow = Hs + (m0 + (l & 15)) * HSTR_ + ((l >> 4) << 1);
            const v2f*  bp    = (const v2f*)(Pu + nt * 8192) + l;
            v8f c = {};
#pragma unroll 4
            for (int kq = 0; kq < 128; ++kq) {
                v2f a = *(const v2f*)(arow + kq * 4);
                v2f b = bp[kq * 32];
                c = __builtin_amdgcn_wmma_f32_16x16x4_f32(false, a, false, b, (short)0, c,
                                                          false, false);
            }
            int n     = nt * 16 + (l & 15);
            int mbase = m0 + ((l >> 4) << 3);
#pragma unroll
            for (int v = 0; v < 8; ++v)
                Ss[(mbase + v) * SSTR_ + n] = c[v];
        }
        __syncthreads();

        // ---- Row phase: 8 lanes per row ---------------------------------
        const float* xsrow = XS + (t * B_ + row) * NSUM_;

        // layernorm stats for xt@W (global XS) and h_tm1@U (LDS S)
        float sx = 0.f, qx = 0.f, sh = 0.f, qh = 0.f;
#pragma unroll 8
        for (int i = 0; i < 32; ++i) {
            int c = i * 8 + j;
            float vx = xsrow[c];
            float vh = Ss[row * SSTR_ + c];
            sx += vx; qx += vx * vx;
            sh += vh; qh += vh * vh;
        }
        for (int d = 1; d < 8; d <<= 1) {
            sx += __shfl_xor(sx, d, 8);
            qx += __shfl_xor(qx, d, 8);
            sh += __shfl_xor(sh, d, 8);
            qh += __shfl_xor(qh, d, 8);
        }
        float mux = sx * (1.f / 256.f), muh = sh * (1.f / 256.f);
        float rsx = rsqrtf(qx * (1.f / 256.f) - mux * mux + 1e-5f);
        float rsh = rsqrtf(qh * (1.f / 256.f) - muh * muh + 1e-5f);

        // FK gate: two dot products over 128 cols
        float dR = 0.f, dT = 0.f;
#pragma unroll 4
        for (int i = 0; i < 16; ++i) {
            int c = i * 8 + j;
            float sfk = Ss[row * SSTR_ + 256 + c] + xsrow[256 + c] + bfk1[c];
            float wv  = wfk2[c];
            dR += hsig(sfk + aR[c]) * wv;
            dT += hsig(sfk + aT[c]) * wv;
        }
        for (int d = 1; d < 8; d <<= 1) {
            dR += __shfl_xor(dR, d, 8);
            dT += __shfl_xor(dT, d, 8);
        }
        float r2 = hsig(dR + bfk2[0]);
        float t2 = hsig(dT + bfk2[0]);
        float fk_cand = (r2 <= t2) ? 1.f : 0.f;

        float mt_   = mask[row * SEQ_ + t];
        float mnext = (t + 1 < SEQ_) ? mask[row * SEQ_ + t + 1] : 0.f;
        float m2t   = mt_ * (1.f - mnext);
        float fkp   = (layer == 0) ? 0.f
                                   : ((t + 1 < SEQ_) ? FKprev[(t + 1) * B_ + row] : 0.f);
        float hvp   = (layer == 0) ? 1.f : HVprev[t * B_ + row];
        float hvt   = hvs[row];

        float fk = fkp + (1.f - fkp) * fk_cand;
        if (mask3_t > 0.f) fk = 1.f;
        if (m2t > 0.f)     fk = 0.f;
        float fkc = mt_ * (1.f - mask3_t) * (1.f - m2t) * (1.f - fkp);

        float h_only = hvt * fk * (fkp + (1.f - fkp) * (1.f - hvp));
        float x_only = hvp * (1.f - fkp) * (1.f - fk + fk * (1.f - hvt));
        float both   = (1.f - fkp) * fk * hvt * hvp;
        float hv     = h_only + x_only + both;
        hv = mt_ * hv + (1.f - mt_) * hvt;

        float*       xrow   = X + ((size_t)(t * B_ + row)) * D_;
        float*       allrow = outALL + (((size_t)row * DEPTH_ + layer) * SEQ_ + t) * D_;
        const bool   last   = (layer == DEPTH_ - 1) && (t == SEQ_ - 1);

        // cols 0..255: candidate h_ is zero there
#pragma unroll 8
        for (int i = 0; i < 32; ++i) {
            int c = i * 8 + j;
            float hold = Hs[row * HSTR_ + c];
            float xv   = xrow[c];
            float hn   = h_only * hold + x_only * xv;
            hn = mt_ * hn + (1.f - mt_) * hold;
            Hs[row * HSTR_ + c] = hn;
            xrow[c]   = hn;
            allrow[c] = hn;
        }
        // cols 256..511: candidate h_ = tanh(s)
#pragma unroll 4
        for (int i = 0; i < 32; ++i) {
            int cc = i * 8 + j;           // 0..255 within hidden part
            int c  = 256 + cc;
            float xw  = xsrow[cc];
            float shv = Ss[row * SSTR_ + cc];
            float lx  = (xw - mux) * rsx * gammas[cc] + betas[cc];
            float lh  = (shv - muh) * rsh * gammas[256 + cc] + betas[256 + cc];
            float s   = (1.f - fkp) * lx + fk * lh + bvec[cc];
            float th  = tanhf(s);

            float hold = Hs[row * HSTR_ + c];
            float xv   = xrow[c];
            float hn   = h_only * hold + x_only * xv + both * th;
            hn = mt_ * hn + (1.f - mt_) * hold;
            Hs[row * HSTR_ + c] = hn;
            xrow[c]   = hn;
            allrow[c] = hn;
            if (last) outO[row * 256 + cc] = hn;
        }

        if (j == 0) {
            FKcur[t * B_ + row] = fk;
            HVcur[t * B_ + row] = hv;
            outFK [((size_t)row * DEPTH_ + layer) * SEQ_ + t] = fk;
            outFKC[((size_t)row * DEPTH_ + layer) * SEQ_ + t] = fkc;
            hvs[row] = hv;
        }
        __syncthreads();
    }
}

// ---------------------------------------------------------------------------
extern "C" void kernel_launch(void* const* d_in, const int* in_sizes, int n_in,
                              void* d_out, int out_size, void* d_ws, size_t ws_size,
                              hipStream_t stream) {
    const float* x      = (const float*)d_in[0];
    const float* mask   = (const float*)d_in[1];
    // d_in[2] = bucket_size == SEQ (constant 96)
    const float* W      = (const float*)d_in[3];
    const float* U      = (const float*)d_in[4];
    const float* b      = (const float*)d_in[5];
    const float* WFK1   = (const float*)d_in[6];
    const float* UFK1   = (const float*)d_in[7];
    const float* bFK1   = (const float*)d_in[8];
    const float* aR     = (const float*)d_in[9];
    const float* aT     = (const float*)d_in[10];
    const float* WFK2   = (const float*)d_in[11];
    const float* bFK2   = (const float*)d_in[12];
    const float* gammas = (const float*)d_in[13];
    const float* betas  = (const float*)d_in[14];
    float* out = (float*)d_out;

    float* ws  = (float*)d_ws;
    float* Pu  = ws;                    // 196608
    float* Pw  = Pu + 196608;           // 196608
    float* XS  = Pw + 196608;           // 96*64*384 = 2359296
    float* X   = XS + 2359296;          // 96*64*512 = 3145728
    float* FKa = X + 3145728;           // 6144
    float* FKb = FKa + 6144;
    float* HVa = FKb + 6144;
    float* HVb = HVa + 6144;            // total ~23.7 MB

    pack_weights<<<1536, 256, 0, stream>>>(U, UFK1, W, WFK1, Pu, Pw);
    build_x0<<<12288, 256, 0, stream>>>(x, X, out);

    const size_t smem = (size_t)(B_ * HSTR_ + B_ * SSTR_ + 64) * sizeof(float);
    float* FKp = FKa; float* HVp = HVa;
    float* FKc = FKb; float* HVc = HVb;
    for (int layer = 0; layer < DEPTH_; ++layer) {
        gemm_x<<<1152, 256, 0, stream>>>(X, Pw, XS);
        scan_layer<<<1, 512, smem, stream>>>(
            X, XS, Pu, mask, b, bFK1, aR, aT, WFK2, bFK2, gammas, betas,
            (layer == 0) ? (const float*)nullptr : FKp,
            (layer == 0) ? (const float*)nullptr : HVp,
            FKc, HVc, out, layer);
        float* tp;
        tp = FKp; FKp = FKc; FKc = tp;
        tp = HVp; HVp = HVc; HVc = tp;
    }
}